// GMMConv_8967891714109
// MI455X (gfx1250) — compile-verified
//
#include <hip/hip_runtime.h>
#include <math.h>

#define NNODES 100000
#define NEDGES 1600000
#define INFEAT 64
#define KMIX 4
#define OUTFEAT 64
#define KOUT 256   // KMIX * OUTFEAT
#define EPS 1e-15f

typedef __attribute__((ext_vector_type(2))) float v2f;
typedef __attribute__((ext_vector_type(8))) float v8f;

// ---------------------------------------------------------------- utilities
__global__ void zero_f32_kernel(float4* __restrict__ p, int n4) {
    int i = blockIdx.x * blockDim.x + threadIdx.x;
    if (i < n4) p[i] = make_float4(0.f, 0.f, 0.f, 0.f);
}

__global__ void deg_count_kernel(const int* __restrict__ dst,
                                 float* __restrict__ deg, int e_cnt) {
    int e = blockIdx.x * blockDim.x + threadIdx.x;
    if (e < e_cnt) atomicAdd(&deg[dst[e]], 1.0f);
}

__global__ void inv_sqrt_deg_kernel(const float* __restrict__ deg,
                                    float* __restrict__ inv, int n) {
    int i = blockIdx.x * blockDim.x + threadIdx.x;
    if (i < n) {
        float d = deg[i];
        inv[i] = (d > 0.0f) ? rsqrtf(d) : 0.0f;   // deg>=1 when >0, so max(d,1)==d
    }
}

// gp[k*4+0]=mu0, +1=mu1, +2=-0.5/(eps+s0^2), +3=-0.5/(eps+s1^2)
__global__ void gauss_prep_kernel(const float* __restrict__ mu,
                                  const float* __restrict__ sigma,
                                  float* __restrict__ gp) {
    int k = threadIdx.x;
    if (k < KMIX) {
        float s0 = sigma[k * 2 + 0], s1 = sigma[k * 2 + 1];
        gp[k * 4 + 0] = mu[k * 2 + 0];
        gp[k * 4 + 1] = mu[k * 2 + 1];
        gp[k * 4 + 2] = -0.5f / (EPS + s0 * s0);
        gp[k * 4 + 3] = -0.5f / (EPS + s1 * s1);
    }
}

// ------------------------------------------------ f32 WMMA GEMM  C = A*B (+bias)
// A: [NNODES x 64] row-major.  B: [64 x NCOLS] row-major.  C: [NNODES x NCOLS].
// Each wave computes a 16x64 strip of C: 4 N-tiles register-blocked, so one
// A 16x4 fragment feeds 4 x V_WMMA_F32_16X16X4_F32 per k-step (64 wmma/wave).
// B panel is staged in LDS once per block (padded stride NCOLS+1 -> the hi/lo
// half-wave B reads hit distinct banks).
// Layouts (ISA 7.12.2): A 16x4: lane%16=M, VGPR0/1 = K{0,1} (lanes 0-15) /
// K{2,3} (lanes 16-31). B 4x16 mirrors A with N on lanes. C/D: VGPR v ->
// row v + 8*(lane>=16), col = lane%16.
template <int NCOLS, bool ADD_BIAS>
__global__ void wmma_gemm_kernel(const float* __restrict__ A,
                                 const float* __restrict__ B,
                                 const float* __restrict__ bias,
                                 float* __restrict__ C) {
    constexpr int GROUPS = NCOLS / 64;          // 16x64 strips per tile row
    constexpr int LDB    = NCOLS + 1;           // padded LDS row stride
    __shared__ float sB[INFEAT * LDB];

    // ---- stage B panel (64 x NCOLS) into LDS, all threads participate ----
    for (int i = threadIdx.x; i < INFEAT * NCOLS; i += blockDim.x) {
        int k = i / NCOLS, n = i - k * NCOLS;
        sB[k * LDB + n] = B[i];
    }
    __syncthreads();

    int gwave = blockIdx.x * (blockDim.x >> 5) + (threadIdx.x >> 5);
    int tileM = gwave / GROUPS;
    int group = gwave % GROUPS;
    if (tileM < NNODES / 16) {                  // wave-uniform: EXEC all-1 inside
        int lane   = threadIdx.x & 31;
        int lane16 = lane & 15;
        int hi     = lane >> 4;                 // 0 or 1
        int kh     = hi << 1;                   // K sub-offset: 0 or 2
        int m      = tileM * 16 + lane16;
        int nbase  = group * 64;

        v8f c[4];
#pragma unroll
        for (int j = 0; j < 4; ++j) {
            float ci = ADD_BIAS ? bias[nbase + j * 16 + lane16] : 0.0f;
#pragma unroll
            for (int v = 0; v < 8; ++v) c[j][v] = ci;
        }

        const float* arow = A + (size_t)m * INFEAT;
#pragma unroll
        for (int k0 = 0; k0 < INFEAT; k0 += 4) {
            v2f a;
            a.x = arow[k0 + kh];
            a.y = arow[k0 + kh + 1];
            const float* b0 = &sB[(k0 + kh) * LDB + nbase + lane16];
            const float* b1 = &sB[(k0 + kh + 1) * LDB + nbase + lane16];
#pragma unroll
            for (int j = 0; j < 4; ++j) {
                v2f b;
                b.x = b0[j * 16];
                b.y = b1[j * 16];
                // 8 args: (neg_a, A, neg_b, B, c_mod, C, reuse_a, reuse_b)
                c[j] = __builtin_amdgcn_wmma_f32_16x16x4_f32(false, a, false, b,
                                                             (short)0, c[j],
                                                             false, false);
            }
        }

        int mbase = tileM * 16 + hi * 8;
#pragma unroll
        for (int j = 0; j < 4; ++j)
#pragma unroll
            for (int v = 0; v < 8; ++v)
                C[(size_t)(mbase + v) * NCOLS + nbase + j * 16 + lane16] = c[j][v];
    }
}

// ------------------------------------------------ per-edge mixture + scatter
// 64 threads per edge: thread m computes msg[m] = sum_k gauss[k]*h[src][k*64+m]
// and atomically accumulates into agg[dst*64+m]. h gathers are coalesced and
// L2-resident (h = 102.4 MB < 192 MB L2).
__global__ void edge_msg_kernel(const int* __restrict__ ei,      // [2*E]: src then dst
                                const float* __restrict__ inv,   // [N] inv_sqrt_deg
                                const float* __restrict__ h,     // [N x 256]
                                const float* __restrict__ gp,    // [4 x 4] prepped
                                float* __restrict__ agg,         // [N x 64]
                                int e_cnt) {
    int t = blockIdx.x * blockDim.x + threadIdx.x;
    int e = t >> 6;
    int m = t & 63;
    if (e >= e_cnt) return;

    int src = ei[e];
    int dst = ei[NEDGES + e];
    float ea0 = inv[src];
    float ea1 = inv[dst];

    const float* hrow = h + (size_t)src * KOUT;
    float msg = 0.0f;
#pragma unroll
    for (int k = 0; k < KMIX; ++k) {
        float d0 = ea0 - gp[k * 4 + 0];
        float d1 = ea1 - gp[k * 4 + 1];
        float arg = fmaf(d0 * d0, gp[k * 4 + 2], d1 * d1 * gp[k * 4 + 3]);
        float w = __expf(arg);
        msg = fmaf(w, hrow[k * OUTFEAT + m], msg);
    }
    atomicAdd(&agg[(size_t)dst * OUTFEAT + m], msg);
}

// ------------------------------------------------ out += agg / max(deg,1)
__global__ void finalize_kernel(const float* __restrict__ agg,
                                const float* __restrict__ deg,
                                float* __restrict__ out, int total) {
    int i = blockIdx.x * blockDim.x + threadIdx.x;
    if (i < total) {
        float d = fmaxf(deg[i >> 6], 1.0f);
        out[i] += agg[i] / d;
    }
}

// ---------------------------------------------------------------- launcher
extern "C" void kernel_launch(void* const* d_in, const int* in_sizes, int n_in,
                              void* d_out, int out_size, void* d_ws, size_t ws_size,
                              hipStream_t stream) {
    const float* x      = (const float*)d_in[0];
    const int*   ei     = (const int*)d_in[1];   // [2, E]
    const float* g      = (const float*)d_in[2]; // [64, 256]
    const float* mu     = (const float*)d_in[3]; // [4, 2]
    const float* sigma  = (const float*)d_in[4]; // [4, 2]
    const float* root_w = (const float*)d_in[5]; // [64, 64]
    const float* bias   = (const float*)d_in[6]; // [64]
    float* out = (float*)d_out;

    // workspace layout (floats): deg[N] | inv[N] | agg[64N] | h[256N] | gp[16]
    float* ws  = (float*)d_ws;
    float* deg = ws;
    float* inv = ws + NNODES;
    float* agg = ws + 2 * (size_t)NNODES;
    float* h   = ws + 66 * (size_t)NNODES;
    float* gp  = ws + 322 * (size_t)NNODES;

    // zero deg+inv+agg region (66N floats) each call -> deterministic replay
    int zcnt4 = 66 * NNODES / 4;
    zero_f32_kernel<<<(zcnt4 + 255) / 256, 256, 0, stream>>>((float4*)ws, zcnt4);

    gauss_prep_kernel<<<1, 32, 0, stream>>>(mu, sigma, gp);
    deg_count_kernel<<<(NEDGES + 255) / 256, 256, 0, stream>>>(ei + NEDGES, deg, NEDGES);
    inv_sqrt_deg_kernel<<<(NNODES + 255) / 256, 256, 0, stream>>>(deg, inv, NNODES);

    // h = x @ g : 6250 tile rows x 4 strips = 25000 waves, 8 waves/block
    wmma_gemm_kernel<KOUT, false><<<(NNODES / 16) * (KOUT / 64) / 8, 256, 0, stream>>>(
        x, g, nullptr, h);

    // out = x @ root_w + bias : 6250 waves (one 16x64 strip each)
    wmma_gemm_kernel<OUTFEAT, true><<<(NNODES / 16 + 7) / 8, 256, 0, stream>>>(
        x, root_w, bias, out);

    // per-edge mixture messages + scatter-add (64 threads per edge)
    long long etotal = (long long)NEDGES * 64;
    edge_msg_kernel<<<(int)(etotal / 256), 256, 0, stream>>>(ei, inv, h, gp,
                                                             agg, NEDGES);

    // out += agg / max(deg,1)
    int total = NNODES * OUTFEAT;
    finalize_kernel<<<(total + 255) / 256, 256, 0, stream>>>(agg, deg, out, total);
}